// TimeDependentRatchetPolicyRIM_14035953123841
// MI455X (gfx1250) — compile-verified
//
#include <hip/hip_runtime.h>

// Rank-1 outer-product via V_WMMA_F32_16X16X4_F32 (CDNA5 / gfx1250, wave32).
typedef __attribute__((ext_vector_type(2))) float v2f;
typedef __attribute__((ext_vector_type(8))) float v8f;

// Problem constants from the reference
#define D_COLS 16
#define G_GRID 2048
#define OUT_STRIDE 17   // D_COLS + 1 (C column)

__global__ __launch_bounds__(256) void ratchet_policy_kernel(
    const float* __restrict__ X,        // (B,2): wealth, habit interleaved
    const float* __restrict__ TmT,      // (B,)
    const float* __restrict__ Y,        // (B,)
    const float* __restrict__ u_star,   // (16,)
    const float* __restrict__ z_tilde,  // (2048,)
    float* __restrict__ out)            // (B,17)
{
    __shared__ float zsh[G_GRID];

    const int t = threadIdx.x;

    // Stage the 8KB interp table into LDS (coalesced, 8 dwords/thread)
    #pragma unroll
    for (int k = 0; k < G_GRID / 256; ++k)
        zsh[t + k * 256] = z_tilde[t + k * 256];
    __syncthreads();

    // ---- per-row scalar pipeline (one row per lane) ----
    const int row = blockIdx.x * 256 + t;          // B divisible by 256 -> no tail
    const float2 xh = ((const float2*)X)[row];     // global_load_b64
    const float wealth = xh.x;
    const float habit  = xh.y;
    const float tau    = TmT[row];
    const float yv     = Y[row];

    // jnp.interp on uniform grid [0,1] with G=2048 points
    float s   = fminf(fmaxf(tau, 0.0f), 1.0f);
    float pos = s * (float)(G_GRID - 1);
    int   i0  = (int)pos;
    i0 = (i0 > G_GRID - 2) ? (G_GRID - 2) : i0;
    float f   = pos - (float)i0;
    float z0  = zsh[i0];
    float z1  = zsh[i0 + 1];
    float z   = fmaf(f, z1 - z0, z0);

    // boundary consumption, ratchet, annuity, drawdown share
    float y_eff = fmaxf(yv, 1e-12f);
    float ratio = fmaxf(z / y_eff, 1e-24f);
    float c_ob  = cbrtf(ratio);                    // GAMMA = 3 -> x^(1/3)
    float h3    = habit * habit * habit;
    bool  need  = (y_eff * h3 <= z);               // need_raise
    float C     = need ? fmaxf(c_ob, habit) : habit;  // ALPHA_RAISE = 1
    float ann   = (1.0f - __expf(-0.02f * tau)) * 50.0f;  // (1-e^{-R tau})/R
    float pvC   = ann * C;
    float dw    = (wealth - pvC) / fmaxf(wealth, 1e-12f);
    dw = fminf(fmaxf(dw, 0.0f), 1.0f);             // clip to [DW_FLOOR, 1]

    // ---- u = dw ⊗ u_star via two 16x16x4 f32 WMMAs per wave ----
    const int lane     = t & 31;
    const int waveBase = blockIdx.x * 256 + (t & ~31);  // first of this wave's 32 rows
    const bool lo      = (lane < 16);

    // lanes 0..15 fetch dw of lanes 16..31 for the second 16-row tile
    float dw_hi = __shfl(dw, lane | 16, 32);

    // Unconditional load (all lanes hit one 64B line) + select: no EXEC branch.
    float us_all = u_star[lane & 15];
    float us     = lo ? us_all : 0.0f;

    // A (16x4): lane m<16 holds K=0 in .x; K=1..3 zero. B (4x16): K=0 row = u_star.
    v2f a0; a0.x = lo ? dw    : 0.0f; a0.y = 0.0f;
    v2f a1; a1.x = lo ? dw_hi : 0.0f; a1.y = 0.0f;
    v2f bm; bm.x = us;                bm.y = 0.0f;
    v8f cz = {0.0f, 0.0f, 0.0f, 0.0f, 0.0f, 0.0f, 0.0f, 0.0f};

    // D[m][n] = dw[m] * u_star[n]
    v8f d0 = __builtin_amdgcn_wmma_f32_16x16x4_f32(false, a0, false, bm,
                                                   (short)0, cz, false, false);
    v8f d1 = __builtin_amdgcn_wmma_f32_16x16x4_f32(false, a1, false, bm,
                                                   (short)0, cz, false, false);

    // D layout: lanes 0-15 -> (M=r, N=lane); lanes 16-31 -> (M=8+r, N=lane-16).
    // Streaming output -> non-temporal stores (TH=NT), two 64B runs per instr.
    const int n    = lane & 15;
    const int mOff = (lane >> 4) << 3;  // 0 or 8
    #pragma unroll
    for (int r = 0; r < 8; ++r) {
        const int m = r + mOff;
        __builtin_nontemporal_store(d0[r],
            out + (size_t)(waveBase + m)      * OUT_STRIDE + n);
        __builtin_nontemporal_store(d1[r],
            out + (size_t)(waveBase + 16 + m) * OUT_STRIDE + n);
    }
    // C column (index 16) for this lane's own row
    __builtin_nontemporal_store(C,
        out + (size_t)(waveBase + lane) * OUT_STRIDE + D_COLS);
}

extern "C" void kernel_launch(void* const* d_in, const int* in_sizes, int n_in,
                              void* d_out, int out_size, void* d_ws, size_t ws_size,
                              hipStream_t stream) {
    // setup_inputs order: X, TmT, y, u_star, s_grid, z_tilde
    const float* X       = (const float*)d_in[0];
    const float* TmT     = (const float*)d_in[1];
    const float* Y       = (const float*)d_in[2];
    const float* u_star  = (const float*)d_in[3];
    // d_in[4] = s_grid: uniform linspace(0,1,G) -> handled analytically
    const float* z_tilde = (const float*)d_in[5];

    const int B = in_sizes[1];        // TmT has B elements; B = 2^21, 256 | B
    const int grid = B / 256;

    ratchet_policy_kernel<<<grid, 256, 0, stream>>>(X, TmT, Y, u_star, z_tilde,
                                                    (float*)d_out);
}